// TrueQuantizationLayer_86079734546646
// MI455X (gfx1250) — compile-verified
//
#include <hip/hip_runtime.h>
#include <stdint.h>

// ---------------------------------------------------------------------------
// SAR-ADC bit-serial quantizer, NBITS=4, NW=10.
//   q  : [LENGTH, NADC]          -> d_out[0 .. n)
//   Q  : [LENGTH, NADC, NBITS]   -> d_out[n .. 5n)
// Pure streaming / HBM-bound problem (288 MB @ 23.3 TB/s ~= 12.4 us floor).
// Strategy: b128 loads via async global->LDS staging (ASYNCcnt path, double
// buffered), ~25 VALU flops per element, b128 non-temporal stores.
// ---------------------------------------------------------------------------

typedef float f4 __attribute__((ext_vector_type(4)));
typedef int v4i __attribute__((ext_vector_type(4)));

#define TPB 256  // 8 wave32 per block

#if defined(__gfx1250__) && __has_builtin(__builtin_amdgcn_global_load_async_to_lds_b128)
#define HAVE_ASYNC_LDS 1
#else
#define HAVE_ASYNC_LDS 0
#endif

#if defined(__gfx1250__) && __has_builtin(__builtin_amdgcn_s_wait_asynccnt)
#define WAIT_ASYNC(n) __builtin_amdgcn_s_wait_asynccnt(n)
#elif defined(__gfx1250__)
#define WAIT_ASYNC(n) asm volatile("s_wait_asynccnt %0" ::"i"(n))
#else
#define WAIT_ASYNC(n)
#endif

__device__ __forceinline__ void async_copy16(const float* g, float* l) {
#if HAVE_ASYNC_LDS
  // Param 1 expects a generic pointer to a 16-byte int vector (per hipcc
  // diagnostic); keep (global_src, lds_dst, imm_offset, imm_cpol) order.
  __builtin_amdgcn_global_load_async_to_lds_b128(
      (v4i*)const_cast<float*>(g), (v4i*)l,
      /*offset=*/0, /*cpol=*/0);
#else
  *(f4*)l = *(const f4*)g;  // fallback: plain b128 load + ds store
#endif
}

__device__ __forceinline__ f4 splat4(float a) {
  f4 v = {a, a, a, a};
  return v;
}

// jnp.sign semantics: -1 / 0 / +1
__device__ __forceinline__ float sgn1(float t) {
  return (t > 0.0f) ? 1.0f : ((t < 0.0f) ? -1.0f : 0.0f);
}
__device__ __forceinline__ f4 sgn4(f4 t) {
  f4 r;
  r.x = sgn1(t.x);
  r.y = sgn1(t.y);
  r.z = sgn1(t.z);
  r.w = sgn1(t.w);
  return r;
}

__global__ __launch_bounds__(TPB) void sar_adc_stream(
    const float* __restrict__ x, const float* __restrict__ W,
    float* __restrict__ q_out, float* __restrict__ Q_out,
    int nv4, int ntiles) {
  __shared__ float lds[2][TPB * 4];  // 2 x 4KB double buffer

  const int tid = threadIdx.x;

  // Uniform scalar weights (compiler emits s_load; broadcast across wave32).
  const float w0 = W[0], w1 = W[1], w2 = W[2], w3 = W[3], w4 = W[4];
  const float w5 = W[5], w6 = W[6], w7 = W[7], w8 = W[8], w9 = W[9];

  const float VDD = 1.8f;
  const float VR = VDD / 16.0f;  // 0.1125
  const float VREF = VR;
  const float DELTA = 1e-30f;

  const int tile0 = blockIdx.x;
  const int stride = gridDim.x;

  // ---- prologue: async-load first tile into buffer 0 ----
  {
    int g = tile0 * TPB + tid;
    g = (g < nv4) ? g : (nv4 - 1);  // clamp: no OOB, no EXEC divergence on DMA
    async_copy16(x + 4 * (size_t)g, &lds[0][tid * 4]);
  }

  int p = 0;
  for (int tile = tile0; tile < ntiles; tile += stride, p ^= 1) {
    const int nexttile = tile + stride;
    if (nexttile < ntiles) {
      int gn = nexttile * TPB + tid;
      gn = (gn < nv4) ? gn : (nv4 - 1);
      async_copy16(x + 4 * (size_t)gn, &lds[p ^ 1][tid * 4]);
      WAIT_ASYNC(1);  // current tile's DMA done; next tile still in flight
    } else {
      WAIT_ASYNC(0);
    }
    asm volatile("" ::: "memory");  // compiler fence: LDS reads after wait

    const f4 xv = *(const f4*)&lds[p][tid * 4];

    // ---- SAR recurrence, weight index m statically unrolled (m = 9..0) ----
    // j = 3
    f4 bs = splat4(w9 * VREF);
    const f4 s3 = sgn4(xv - bs + DELTA);
    // j = 2
    bs = splat4(w8 * VREF);
    bs = bs + ((s3 + 1.0f) * 0.5f) * w7 * VR;
    const f4 s2 = sgn4(xv - bs + DELTA);
    // j = 1
    bs = splat4(w6 * VREF);
    bs = bs + ((s2 + 1.0f) * 0.5f) * w5 * VR;
    bs = bs + ((s3 + 1.0f) * 0.5f) * w4 * VR;
    const f4 s1 = sgn4(xv - bs + DELTA);
    // j = 0
    bs = splat4(w3 * VREF);
    bs = bs + ((s1 + 1.0f) * 0.5f) * w2 * VR;
    bs = bs + ((s2 + 1.0f) * 0.5f) * w1 * VR;
    bs = bs + ((s3 + 1.0f) * 0.5f) * w0 * VR;
    const f4 s0 = sgn4(xv - bs + DELTA);

    // q = sum_j (Q_j + 1) * 0.5 * VR * 2^j
    const f4 q = ((s0 + 1.0f) * 0.5f) * (VR * 1.0f) +
                 ((s1 + 1.0f) * 0.5f) * (VR * 2.0f) +
                 ((s2 + 1.0f) * 0.5f) * (VR * 4.0f) +
                 ((s3 + 1.0f) * 0.5f) * (VR * 8.0f);

    const int g = tile * TPB + tid;
    if (g < nv4) {
      // Transpose 4 elements x 4 bits -> 4 contiguous b128 rows of Q.
      const f4 o0 = {s0.x, s1.x, s2.x, s3.x};
      const f4 o1 = {s0.y, s1.y, s2.y, s3.y};
      const f4 o2 = {s0.z, s1.z, s2.z, s3.z};
      const f4 o3 = {s0.w, s1.w, s2.w, s3.w};

      __builtin_nontemporal_store(q, (f4*)(q_out + 4 * (size_t)g));
      const size_t qb = 16 * (size_t)g;
      __builtin_nontemporal_store(o0, (f4*)(Q_out + qb + 0));
      __builtin_nontemporal_store(o1, (f4*)(Q_out + qb + 4));
      __builtin_nontemporal_store(o2, (f4*)(Q_out + qb + 8));
      __builtin_nontemporal_store(o3, (f4*)(Q_out + qb + 12));
    }
  }
}

extern "C" void kernel_launch(void* const* d_in, const int* in_sizes, int n_in,
                              void* d_out, int out_size, void* d_ws, size_t ws_size,
                              hipStream_t stream) {
  (void)n_in;
  (void)out_size;
  (void)d_ws;
  (void)ws_size;

  const float* x = (const float*)d_in[0];  // [LENGTH*NADC] = 12,000,000 f32
  const float* W = (const float*)d_in[1];  // [10] f32
  float* out = (float*)d_out;

  const int n = in_sizes[0];  // 12,000,000 (divisible by 4)
  const int nv4 = n / 4;
  const int ntiles = (nv4 + TPB - 1) / TPB;
  int blocks = ntiles < 4096 ? ntiles : 4096;

  sar_adc_stream<<<blocks, TPB, 0, stream>>>(x, W, out, out + n, nv4, ntiles);
}